// NasAutoGraphDCell_87402584473615
// MI455X (gfx1250) — compile-verified
//
#include <hip/hip_runtime.h>
#include <math.h>

// ---------------------------------------------------------------------------
// MI455X (gfx1250) implementation of the NAS graph cell.
// fp32 throughout (op is bandwidth/L2-atomic bound, not matrix-pipe bound),
// dense GEMMs run on the matrix pipe via V_WMMA_F32_16X16X4_F32.
// ---------------------------------------------------------------------------

#define NN 50000
#define NE 800000
#define DD 128

typedef __attribute__((ext_vector_type(2))) float v2f;
typedef __attribute__((ext_vector_type(8))) float v8f;

// ------------------------- degree / norm kernels ---------------------------

__global__ void k_degree(const int* __restrict__ col, const float* __restrict__ w,
                         float* __restrict__ deg, int E) {
    int e = blockIdx.x * blockDim.x + threadIdx.x;
    if (e < E) atomicAdd(&deg[col[e]], w[e]);
}

__global__ void k_dinv(const float* __restrict__ deg,
                       float* __restrict__ dsl, float* __restrict__ dns, int n) {
    int i = blockIdx.x * blockDim.x + threadIdx.x;
    if (i < n) {
        float d = deg[i];
        dsl[i] = rsqrtf(d + 1.0f);            // with self-loop (weight 1.0 added)
        dns[i] = d > 0.0f ? rsqrtf(d) : 0.0f; // without self-loop
    }
}

__global__ void k_edgenorm(const int* __restrict__ row, const int* __restrict__ col,
                           const float* __restrict__ w,
                           const float* __restrict__ dsl, const float* __restrict__ dns,
                           float* __restrict__ nsl, float* __restrict__ nns, int E) {
    int e = blockIdx.x * blockDim.x + threadIdx.x;
    if (e < E) {
        int r = row[e], c = col[e];
        float we = w[e];
        nsl[e] = dsl[r] * we * dsl[c];
        nns[e] = dns[r] * we * dns[c];
    }
}

// ------------------------------ scatter ------------------------------------
// out[col] += norm * feat[row] for both propagations. Thread per (edge, dim).
// Working set (~102MB) fits MI455X's 192MB L2, so atomics resolve in L2.

__global__ void k_scatter(const int* __restrict__ row, const int* __restrict__ col,
                          const float* __restrict__ nsl, const float* __restrict__ nns,
                          const float* __restrict__ hp, const float* __restrict__ ai,
                          float* __restrict__ sgp, float* __restrict__ arp, int E) {
    int gid = blockIdx.x * blockDim.x + threadIdx.x;
    int e = gid >> 7;        // / 128
    int d = gid & 127;       // % 128
    if (e < E) {
        int r = row[e], c = col[e];
        atomicAdd(&sgp[c * DD + d], nsl[e] * hp[r * DD + d]);
        atomicAdd(&arp[c * DD + d], nns[e] * ai[r * DD + d]);
    }
}

// self-loop contribution for the SGConv path: sgp[i] += dsl[i]^2 * hp[i]
__global__ void k_selfloop(const float* __restrict__ hp, const float* __restrict__ dsl,
                           float* __restrict__ sgp, int total) {
    int gid = blockIdx.x * blockDim.x + threadIdx.x;
    if (gid < total) {
        int i = gid >> 7;
        float s = dsl[i];
        sgp[gid] += s * s * hp[gid];
    }
}

// ------------------------------ WMMA GEMM ----------------------------------
// out[M=N_nodes x 128] = act( A[Mx128] @ W[128x128] (+ Cpre) + bias )
// One 16x16 output tile per wave; 8 waves/block cover all 128 columns;
// grid.x = rows/16 (50000/16 = 3125 exactly). K loop: 32 x wmma_f32_16x16x4.
//
// ACT: 0 = none
//      1 = elu(leaky_relu(x, 0.01))  ==  x > 0 ? x : exp(0.01x) - 1   (SGConv)
//      2 = relu (ARMA path; subsequent leaky+elu are identity on >=0)
// Template specialization keeps the epilogue branch-free.

template <int ACT, bool HAS_BIAS, bool HAS_CPRE>
__global__ void __launch_bounds__(256)
k_gemm128(const float* __restrict__ A, const float* __restrict__ W,
          const float* __restrict__ bias, const float* __restrict__ Cpre,
          float* __restrict__ out, int ldOut, int colOff) {
    const int wave = threadIdx.x >> 5;
    const int lane = threadIdx.x & 31;
    const int half = lane >> 4;   // 0: K=k0..k0+1 | 1: K=k0+2..k0+3 (A/B frag split)
    const int lo   = lane & 15;
    const int rowBase = blockIdx.x * 16;
    const int col     = wave * 16 + lo;

    v8f c = {};
    if (HAS_CPRE) {
        #pragma unroll
        for (int v = 0; v < 8; ++v) {
            int m = rowBase + v + 8 * half;   // C/D layout: VGPR v <-> M = v + 8*half
            c[v] = Cpre[m * DD + col];
        }
    }

    const float* arow = A + (rowBase + lo) * DD;
    #pragma unroll 8
    for (int k0 = 0; k0 < DD; k0 += 4) {
        const int ka = k0 + 2 * half;
        v2f a;
        a.x = arow[ka];
        a.y = arow[ka + 1];
        v2f b;
        b.x = W[ka * DD + col];
        b.y = W[(ka + 1) * DD + col];
        // D = A(16x4, f32) x B(4x16, f32) + C(16x16, f32)
        c = __builtin_amdgcn_wmma_f32_16x16x4_f32(
                /*neg_a=*/false, a, /*neg_b=*/false, b,
                /*c_mod=*/(short)0, c, /*reuse_a=*/false, /*reuse_b=*/false);
    }

    const float bv = HAS_BIAS ? bias[col] : 0.0f;
    #pragma unroll
    for (int v = 0; v < 8; ++v) {
        int m = rowBase + v + 8 * half;
        float x = c[v] + bv;
        if (ACT == 1) {
            // elu(leaky_relu(x)): for x<=0 both fire -> exp(0.01*x)-1
            x = x > 0.0f ? x : (__expf(0.01f * x) - 1.0f);
        } else if (ACT == 2) {
            x = x > 0.0f ? x : 0.0f;
        }
        out[m * ldOut + colOff + col] = x;
    }
}

// ------------------------------ launcher -----------------------------------

extern "C" void kernel_launch(void* const* d_in, const int* in_sizes, int n_in,
                              void* d_out, int out_size, void* d_ws, size_t ws_size,
                              hipStream_t stream) {
    const float* h   = (const float*)d_in[0];
    const float* x   = (const float*)d_in[1];
    const int*   ei  = (const int*)  d_in[2];   // (2, E): row = ei[0:E], col = ei[E:2E]
    const float* ew  = (const float*)d_in[3];
    const float* Wh  = (const float*)d_in[4];
    const float* bh  = (const float*)d_in[5];
    const float* Wx  = (const float*)d_in[6];
    const float* bx  = (const float*)d_in[7];
    const float* Wsg = (const float*)d_in[8];
    const float* bsg = (const float*)d_in[9];
    const float* Wai = (const float*)d_in[10];  // ARMA init_weight
    const float* War = (const float*)d_in[11];  // ARMA root_weight
    const float* ba  = (const float*)d_in[12];

    float* out = (float*)d_out;
    const int* erow = ei;
    const int* ecol = ei + NE;

    // workspace layout
    char* ws = (char*)d_ws;
    float* deg = (float*)ws;  ws += (size_t)NN * 4;
    float* dsl = (float*)ws;  ws += (size_t)NN * 4;
    float* dns = (float*)ws;  ws += (size_t)NN * 4;
    float* nsl = (float*)ws;  ws += (size_t)NE * 4;
    float* nns = (float*)ws;  ws += (size_t)NE * 4;
    float* hp  = (float*)ws;  ws += (size_t)NN * DD * 4;
    float* xp  = (float*)ws;  ws += (size_t)NN * DD * 4;
    float* ai  = (float*)ws;  ws += (size_t)NN * DD * 4;
    float* sgp = (float*)ws;  ws += (size_t)NN * DD * 4;
    float* arp = (float*)ws;  ws += (size_t)NN * DD * 4;

    // accumulators must be re-zeroed every replay (atomic accumulation)
    hipMemsetAsync(deg, 0, (size_t)NN * 4, stream);
    hipMemsetAsync(sgp, 0, (size_t)NN * DD * 4, stream);
    hipMemsetAsync(arp, 0, (size_t)NN * DD * 4, stream);

    const int TB = 256;
    // degree / dinv / per-edge norms
    k_degree<<<(NE + TB - 1) / TB, TB, 0, stream>>>(ecol, ew, deg, NE);
    k_dinv<<<(NN + TB - 1) / TB, TB, 0, stream>>>(deg, dsl, dns, NN);
    k_edgenorm<<<(NE + TB - 1) / TB, TB, 0, stream>>>(erow, ecol, ew, dsl, dns, nsl, nns, NE);

    // dense pre-GEMMs on the matrix pipe (fp32 WMMA)
    const int RB = NN / 16;  // 3125 row-blocks, 8 waves/block = full 128 cols
    k_gemm128<0, true,  false><<<RB, TB, 0, stream>>>(h,  Wh,  bh,      nullptr, hp, DD, 0);
    k_gemm128<0, true,  false><<<RB, TB, 0, stream>>>(x,  Wx,  bx,      nullptr, xp, DD, 0);
    k_gemm128<0, false, false><<<RB, TB, 0, stream>>>(xp, Wai, nullptr, nullptr, ai, DD, 0);

    // both propagations in one scatter pass (L2-resident working set)
    k_scatter<<<(NE * DD) / TB, TB, 0, stream>>>(erow, ecol, nsl, nns, hp, ai, sgp, arp, NE);
    k_selfloop<<<(NN * DD + TB - 1) / TB, TB, 0, stream>>>(hp, dsl, sgp, NN * DD);

    // fused output GEMMs: write elu(concat(o1,o2)) directly into out[N, 256]
    float* o3 = out + (size_t)NN * DD;
    k_gemm128<1, true, false><<<RB, TB, 0, stream>>>(sgp, Wsg, bsg, nullptr, o3, 2 * DD, 0);  // o1
    k_gemm128<2, true, true ><<<RB, TB, 0, stream>>>(xp,  War, ba,  arp,     o3, 2 * DD, DD); // o2

    // his = x
    hipMemcpyAsync(out, x, (size_t)NN * DD * 4, hipMemcpyDeviceToDevice, stream);
}